// GATLayer_11553462026822
// MI455X (gfx1250) — compile-verified
//
#include <hip/hip_runtime.h>

#define IN_DIM  256
#define OUT_DIM 64
#define N_HEADS 4
#define GAT_EPS 1e-12f

typedef float v2f __attribute__((ext_vector_type(2)));
typedef float v8f __attribute__((ext_vector_type(8)));

// ---------------------------------------------------------------------------
// Pass 1: zero the output accumulator (harness poisons it with 0xAA).
// ---------------------------------------------------------------------------
__global__ __launch_bounds__(256) void gat_zero(float4* __restrict__ p, long n4) {
    long i = (long)blockIdx.x * blockDim.x + threadIdx.x;
    if (i < n4) p[i] = make_float4(0.f, 0.f, 0.f, 0.f);
}

// ---------------------------------------------------------------------------
// Pass 2: x = h @ W via V_WMMA_F32_16X16X4_F32 (fp32 WMMA, wave32).
// Block = 128 threads = 4 waves; block handles 16 rows, wave w handles
// columns [16w, 16w+16). K loop: 256 / 4 = 64 WMMAs per wave (unrolls 8x
// into clause-batched loads + back-to-back v_wmma).
//
// VGPR layouts per CDNA5 ISA §7.12.2:
//   A 16x4 : lanes 0-15 -> M=lane, {VGPR0,VGPR1} = K,K+1 with K = 2*(lane>>4)
//   B 4x16 : lanes 0-15 -> N=lane, mirrored K split across lane halves
//   C/D    : VGPR r, lanes 0-15 -> (M=r, N=lane); lanes 16-31 -> (M=r+8, N=lane-16)
// ---------------------------------------------------------------------------
__global__ __launch_bounds__(128) void gat_gemm_wmma(
    const float* __restrict__ h, const float* __restrict__ W,
    float* __restrict__ x, int N)
{
    const int lane = threadIdx.x & 31;
    const int wave = threadIdx.x >> 5;
    const int half = lane >> 4;      // 0: K,K+1   1: K+2,K+3
    const int l15  = lane & 15;
    const int row0 = blockIdx.x * 16;
    const int col0 = wave * 16;

    int arow = row0 + l15;
    if (arow >= N) arow = N - 1;                 // clamp for a ragged last tile
    const float* __restrict__ hrow = h + (size_t)arow * IN_DIM;
    const int bcol = col0 + l15;

    v8f acc = {0.f, 0.f, 0.f, 0.f, 0.f, 0.f, 0.f, 0.f};

    for (int k = 0; k < IN_DIM; k += 4) {
        const int ka = k + half * 2;
        v2f a = *(const v2f*)(hrow + ka);        // {A[M][ka], A[M][ka+1]}
        v2f b;
        b.x = W[(size_t)ka       * OUT_DIM + bcol];
        b.y = W[(size_t)(ka + 1) * OUT_DIM + bcol];
        acc = __builtin_amdgcn_wmma_f32_16x16x4_f32(
            /*neg_a=*/false, a, /*neg_b=*/false, b,
            /*c_mod=*/(short)0, acc, /*reuse_a=*/false, /*reuse_b=*/false);
    }

    // Epilogue: full tiles (the common case) store unguarded -> 8 straight
    // global_store_b32 per lane, no exec-mask churn.
    float* __restrict__ xout =
        x + ((size_t)(row0 + half * 8)) * OUT_DIM + col0 + l15;
    if (row0 + 16 <= N) {
        #pragma unroll
        for (int r = 0; r < 8; ++r)
            xout[(size_t)r * OUT_DIM] = acc[r];
    } else {
        #pragma unroll
        for (int r = 0; r < 8; ++r) {
            if (row0 + r + half * 8 < N)
                xout[(size_t)r * OUT_DIM] = acc[r];
        }
    }
}

// ---------------------------------------------------------------------------
// Pass 3: el = x @ Wl^T, er = x @ Wr^T  (tiny: N x 64 x 4), zero rowsum.
// ---------------------------------------------------------------------------
__global__ __launch_bounds__(256) void gat_coef(
    const float* __restrict__ x, const float* __restrict__ Wl,
    const float* __restrict__ Wr, float* __restrict__ el,
    float* __restrict__ er, float* __restrict__ rowsum, int N)
{
    const int n = blockIdx.x * blockDim.x + threadIdx.x;
    if (n >= N) return;
    const float* __restrict__ xr = x + (size_t)n * OUT_DIM;
    float al[N_HEADS] = {0.f, 0.f, 0.f, 0.f};
    float ar[N_HEADS] = {0.f, 0.f, 0.f, 0.f};
    for (int d = 0; d < OUT_DIM; ++d) {
        const float xv = xr[d];
        #pragma unroll
        for (int hh = 0; hh < N_HEADS; ++hh) {
            al[hh] = fmaf(xv, Wl[hh * OUT_DIM + d], al[hh]);
            ar[hh] = fmaf(xv, Wr[hh * OUT_DIM + d], ar[hh]);
        }
    }
    #pragma unroll
    for (int hh = 0; hh < N_HEADS; ++hh) {
        el[(size_t)n * N_HEADS + hh] = al[hh];
        er[(size_t)n * N_HEADS + hh] = ar[hh];
        rowsum[(size_t)n * N_HEADS + hh] = 0.f;
    }
}

__device__ __forceinline__ float gat_edge_e(float l, float r) {
    float v = l + r;
    v = (v > 0.f) ? v : 0.2f * v;          // leaky_relu(0.2)
    return __expf(v);
}

// ---------------------------------------------------------------------------
// Pass 4: per-edge unnormalized weights -> atomic rowsum[src][h].
// ---------------------------------------------------------------------------
__global__ __launch_bounds__(256) void gat_rowsum(
    const int* __restrict__ ei, const float* __restrict__ el,
    const float* __restrict__ er, float* __restrict__ rowsum, int E)
{
    const int e = blockIdx.x * blockDim.x + threadIdx.x;
    if (e >= E) return;
    const int src = ei[e];
    const int dst = ei[E + e];
    const float4 l = *(const float4*)(el + (size_t)src * N_HEADS);
    const float4 r = *(const float4*)(er + (size_t)dst * N_HEADS);
    float* rs = rowsum + (size_t)src * N_HEADS;
    atomicAdd(rs + 0, gat_edge_e(l.x, r.x));
    atomicAdd(rs + 1, gat_edge_e(l.y, r.y));
    atomicAdd(rs + 2, gat_edge_e(l.z, r.z));
    atomicAdd(rs + 3, gat_edge_e(l.w, r.w));
}

// ---------------------------------------------------------------------------
// Pass 5: wave-per-edge SpMM scatter. Each lane handles 2 of the 64 dims of
// x[dst] (one coalesced 256 B row load per edge), then 8 f32 atomic adds
// (4 heads x 2 dims) into out[src]. All operands are L2-resident; per wave
// and head, the 32 lanes' atomics cover one contiguous 256 B region.
// ---------------------------------------------------------------------------
__global__ __launch_bounds__(256) void gat_aggregate(
    const int* __restrict__ ei, const float* __restrict__ el,
    const float* __restrict__ er, const float* __restrict__ rowsum,
    const float* __restrict__ x, float* __restrict__ out, int E)
{
    const int e    = blockIdx.x * 8 + (threadIdx.x >> 5);
    const int lane = threadIdx.x & 31;
    if (e >= E) return;
    const int src = ei[e];
    const int dst = ei[E + e];

    const float4 l  = *(const float4*)(el     + (size_t)src * N_HEADS);
    const float4 r  = *(const float4*)(er     + (size_t)dst * N_HEADS);
    const float4 rs = *(const float4*)(rowsum + (size_t)src * N_HEADS);

    float w[N_HEADS];
    w[0] = gat_edge_e(l.x, r.x) / fmaxf(rs.x, GAT_EPS);
    w[1] = gat_edge_e(l.y, r.y) / fmaxf(rs.y, GAT_EPS);
    w[2] = gat_edge_e(l.z, r.z) / fmaxf(rs.z, GAT_EPS);
    w[3] = gat_edge_e(l.w, r.w) / fmaxf(rs.w, GAT_EPS);

    const v2f xv = *(const v2f*)(x + (size_t)dst * OUT_DIM + lane * 2);
    float* obase = out + (size_t)src * (N_HEADS * OUT_DIM) + lane * 2;
    #pragma unroll
    for (int hh = 0; hh < N_HEADS; ++hh) {
        atomicAdd(obase + hh * OUT_DIM + 0, w[hh] * xv.x);
        atomicAdd(obase + hh * OUT_DIM + 1, w[hh] * xv.y);
    }
}

// ---------------------------------------------------------------------------
// Pass 6: out = elu(out + b). Layout [N, H, D] -> bias index = i & 63.
// ---------------------------------------------------------------------------
__global__ __launch_bounds__(256) void gat_finalize(
    float* __restrict__ out, const float* __restrict__ b, long n4)
{
    const long i4 = (long)blockIdx.x * blockDim.x + threadIdx.x;
    if (i4 >= n4) return;
    float4 v = ((const float4*)out)[i4];
    const int d0 = (int)((i4 * 4) & (OUT_DIM - 1));
    v.x += b[d0 + 0]; v.y += b[d0 + 1]; v.z += b[d0 + 2]; v.w += b[d0 + 3];
    v.x = (v.x > 0.f) ? v.x : (__expf(v.x) - 1.f);
    v.y = (v.y > 0.f) ? v.y : (__expf(v.y) - 1.f);
    v.z = (v.z > 0.f) ? v.z : (__expf(v.z) - 1.f);
    v.w = (v.w > 0.f) ? v.w : (__expf(v.w) - 1.f);
    ((float4*)out)[i4] = v;
}

// ---------------------------------------------------------------------------
// Host-side launch. Inputs (setup_inputs order):
//   0: h  [N,256] f32     1: W [256,64] f32    2: Wl [4,64] f32
//   3: Wr [4,64] f32      4: b [64] f32        5: edge_index [2,E] i32
// Output: [N, 256] f32.
// ---------------------------------------------------------------------------
extern "C" void kernel_launch(void* const* d_in, const int* in_sizes, int n_in,
                              void* d_out, int out_size, void* d_ws, size_t ws_size,
                              hipStream_t stream) {
    const float* h  = (const float*)d_in[0];
    const float* W  = (const float*)d_in[1];
    const float* Wl = (const float*)d_in[2];
    const float* Wr = (const float*)d_in[3];
    const float* b  = (const float*)d_in[4];
    const int*   ei = (const int*)d_in[5];

    const int N = in_sizes[0] / IN_DIM;
    const int E = in_sizes[5] / 2;

    float* out = (float*)d_out;

    // Workspace carve-up: x | el | er | rowsum  (~15.2 MB total)
    float* x      = (float*)d_ws;
    float* el     = x  + (size_t)N * OUT_DIM;
    float* er     = el + (size_t)N * N_HEADS;
    float* rowsum = er + (size_t)N * N_HEADS;

    const long out_n4 = (long)out_size / 4;

    gat_zero<<<(unsigned)((out_n4 + 255) / 256), 256, 0, stream>>>(
        (float4*)out, out_n4);

    gat_gemm_wmma<<<(unsigned)((N + 15) / 16), 128, 0, stream>>>(h, W, x, N);

    gat_coef<<<(unsigned)((N + 255) / 256), 256, 0, stream>>>(
        x, Wl, Wr, el, er, rowsum, N);

    gat_rowsum<<<(unsigned)((E + 255) / 256), 256, 0, stream>>>(
        ei, el, er, rowsum, E);

    gat_aggregate<<<(unsigned)((E + 7) / 8), 256, 0, stream>>>(
        ei, el, er, rowsum, x, out, E);

    gat_finalize<<<(unsigned)((out_n4 + 255) / 256), 256, 0, stream>>>(
        out, b, out_n4);
}